// FakeQuantizeV2_74852690034804
// MI455X (gfx1250) — compile-verified
//
#include <hip/hip_runtime.h>
#include <math.h>

// NF5 block fake-quantization for MI455X (gfx1250, wave32).
//
// Two kernels:
//  1) nf5_init_kernel (1 wave): builds an exactly-antisymmetric NF5 grid and
//     decision boundaries via erfinvf into d_ws (interleaved g,b per index).
//  2) FakeQuantizeNF5_kernel<GUARDED>: streaming quantizer. Each lane owns a
//     float4, 8 consecutive lanes own one 32-element block. Grid/boundaries
//     live in registers (lane i <-> index i); searchsorted is a 4-probe
//     lower-bound over the 16 upper boundaries on |xn| via raw
//     ds_bpermute_b32 with byte-scaled indices, sign restored afterwards
//     (exact for the +0 tie case via `xn > 0`). GUARDED=false is used when
//     the launch tiles n exactly (true for 8192x8192) -> no exec-mask
//     save/restore in the unrolled loop.

typedef float v4f __attribute__((ext_vector_type(4)));
typedef float v2f __attribute__((ext_vector_type(2)));

#define NF5_TPB   256   // 8 waves per workgroup
#define NF5_ITERS 8     // float4 chunks per thread (grid-stride)

// -------- init: build grid (g) and boundaries (b), interleaved in tbl --------
__global__ void nf5_init_kernel(float* __restrict__ tbl) {
  const int i = threadIdx.x & 31;
  const float off = 0.9677083f;
  const float p0  = 1.0f - off;
  const float dp  = (off - p0) * (1.0f / 31.0f);
  float p    = p0 + dp * (float)i;
  float graw = 1.41421356237f * erfinvf(2.0f * p - 1.0f);
  // Exact antisymmetrization: gs[31-i] == -gs[i] bitwise.
  float gmir = __shfl(graw, 31 - i, 32);
  float gs   = 0.5f * (graw - gmir);
  float gmax = __shfl(gs, 31, 32);          // positive peak
  float g    = gs / gmax;                    // grid, g[31]=1, g[i]=-g[31-i]
  float gn   = __shfl(g, (i + 1) & 31, 32);
  float b    = (i == 31) ? __builtin_inff() : 0.5f * (g + gn);  // b[15]==0
  tbl[2 * i]     = g;
  tbl[2 * i + 1] = b;
}

// -------- helpers --------
template <int MASK>
__device__ __forceinline__ float swz_xor(float v) {
  // group-of-32 swizzle: and=0x1f, or=0, xor=MASK  -> single ds_swizzle_b32
  return __int_as_float(
      __builtin_amdgcn_ds_swizzle(__float_as_int(v), (MASK << 10) | 0x1f));
}

// searchsorted over upper boundaries on |xn|, sign restored.
// t = 4*idx - 4 (byte-scaled); probes are ds_bpermute_b32 gathers.
__device__ __forceinline__ float nf5_quant1(float xv, float inv, float scale,
                                            int gbits, int bbits) {
  float s = fabsf(xv) * inv;   // |xn|; abs folds into src modifier
  int t = 60;                  // idx = 16
#pragma unroll
  for (int s4 = 32; s4 >= 4; s4 >>= 1) {
    int a = t + s4;            // byte index of boundary[idx + step - 1]
    float bb = __int_as_float(__builtin_amdgcn_ds_bpermute(a, bbits));
    t = (bb < s) ? a : t;
  }
  float gq = __int_as_float(__builtin_amdgcn_ds_bpermute(t + 4, gbits));
  float q  = (xv > 0.0f) ? gq : -gq;  // handles +0 tie like searchsorted
  return q * scale;
}

// -------- main streaming kernel --------
template <bool GUARDED>
__global__ __launch_bounds__(NF5_TPB)
void FakeQuantizeNF5_kernel(const float* __restrict__ x,
                            float* __restrict__ out,
                            const float* __restrict__ tbl,
                            int n4, int stride4) {
  const int lane = threadIdx.x & 31;
  // lane i holds (grid[i], boundary[i]) — one b64 load, L2-broadcast
  v2f gb = ((const v2f*)tbl)[lane];
  const int gbits = __float_as_int(gb.x);
  const int bbits = __float_as_int(gb.y);

  const int tid = blockIdx.x * NF5_TPB + threadIdx.x;

#pragma unroll
  for (int it = 0; it < NF5_ITERS; ++it) {
    const int f   = tid + it * stride4;   // float4 index
    const bool in = GUARDED ? (f < n4) : true;

    v4f v = {0.f, 0.f, 0.f, 0.f};
    if (in) v = __builtin_nontemporal_load((const v4f*)x + f);

    // Early-pull next grid-stride chunk (gfx1250 global_prefetch_b8).
    if (it + 1 < NF5_ITERS) {
      const int fn = tid + (it + 1) * stride4;
      if (!GUARDED || fn < n4) __builtin_prefetch((const v4f*)x + fn, 0, 0);
    }

    // ---- top-2 |x| over the 32-element block (8-lane group, xor 1,2,4) ----
    float h1 = fmaxf(fabsf(v.x), fabsf(v.y)), l1 = fminf(fabsf(v.x), fabsf(v.y));
    float h2 = fmaxf(fabsf(v.z), fabsf(v.w)), l2 = fminf(fabsf(v.z), fabsf(v.w));
    float m1 = fmaxf(h1, h2);                        // local max of 4
    float m2 = fmaxf(fminf(h1, h2), fmaxf(l1, l2));  // local 2nd of 4
    {
      float o1 = swz_xor<1>(m1), o2 = swz_xor<1>(m2);
      float hi = fmaxf(m1, o1), lo = fminf(m1, o1);
      m1 = hi; m2 = fmaxf(lo, fmaxf(m2, o2));
    }
    {
      float o1 = swz_xor<2>(m1), o2 = swz_xor<2>(m2);
      float hi = fmaxf(m1, o1), lo = fminf(m1, o1);
      m1 = hi; m2 = fmaxf(lo, fmaxf(m2, o2));
    }
    {
      float o1 = swz_xor<4>(m1), o2 = swz_xor<4>(m2);
      float hi = fmaxf(m1, o1), lo = fminf(m1, o1);
      m1 = hi; m2 = fmaxf(lo, fmaxf(m2, o2));
    }

    // percentile(99.9, n=32): pos = 31*0.999 = 30.969 -> lerp top-2
    float scale = fmaf(0.969f, m1 - m2, m2);
    scale = fmaxf(scale, 1e-12f);
    float inv = __builtin_amdgcn_rcpf(scale);   // v_rcp_f32, amortized over 4

    v4f r;
    r.x = nf5_quant1(v.x, inv, scale, gbits, bbits);
    r.y = nf5_quant1(v.y, inv, scale, gbits, bbits);
    r.z = nf5_quant1(v.z, inv, scale, gbits, bbits);
    r.w = nf5_quant1(v.w, inv, scale, gbits, bbits);

    if (in) __builtin_nontemporal_store(r, (v4f*)out + f);
  }
}

extern "C" void kernel_launch(void* const* d_in, const int* in_sizes, int n_in,
                              void* d_out, int out_size, void* d_ws, size_t ws_size,
                              hipStream_t stream) {
  const float* x = (const float*)d_in[0];
  float* out     = (float*)d_out;
  float* tbl     = (float*)d_ws;   // 64 floats = 256 B scratch

  const long long n = (long long)in_sizes[0];   // multiple of 32 by construction
  const int n4      = (int)(n / 4);             // float4 count

  nf5_init_kernel<<<dim3(1), dim3(32), 0, stream>>>(tbl);

  const int totalThreads = (n4 + NF5_ITERS - 1) / NF5_ITERS;
  const int blocks       = (totalThreads + NF5_TPB - 1) / NF5_TPB;
  const int stride4      = blocks * NF5_TPB;

  const bool exact = (n4 == blocks * NF5_TPB * NF5_ITERS);  // true for 8192^2
  if (exact) {
    FakeQuantizeNF5_kernel<false><<<dim3(blocks), dim3(NF5_TPB), 0, stream>>>(
        x, out, tbl, n4, stride4);
  } else {
    FakeQuantizeNF5_kernel<true><<<dim3(blocks), dim3(NF5_TPB), 0, stream>>>(
        x, out, tbl, n4, stride4);
  }
}